// KernelEncoderLayer_3813930959351
// MI455X (gfx1250) — compile-verified
//
#include <hip/hip_runtime.h>
#include <hip/hip_bf16.h>
#include <math.h>
#include <stdint.h>

// ---------------------------------------------------------------------------
// Problem constants (fixed by the reference: B=8, n=1024, C=32, Cm=128, Kf=9)
// ---------------------------------------------------------------------------
#define B_    8
#define NPG_  1024          // nodes per graph
#define C_    32
#define CM_   128
#define KF_   9
#define NT_   8192          // total nodes  (B*n)
#define Q_    9216          // n*Kf components per graph
#define KSEL_ 512           // ceil(0.5*n)
#define NEG_SLOPE 0.01f
#define BN_EPS    1e-5f

typedef __attribute__((ext_vector_type(2))) float v2f;
typedef __attribute__((ext_vector_type(4))) float v4f;
typedef __attribute__((ext_vector_type(8))) float v8f;
typedef __attribute__((ext_vector_type(4))) int   v4i;

// ---------------------------------------------------------------------------
// Workspace layout (float offsets).  Total ~4.51M floats (~18 MB).
// ---------------------------------------------------------------------------
#define WS_CONVW   0                       // [B*Q, 32]
#define WS_GEO     2359296                 // [B*Q, 4]  (cx, cy, |q|^2, 0)
#define WS_SP2     2654208                 // [N]
#define WS_X       2662400                 // [N,32]   x after BN1 + residual
#define WS_XL      2924544                 // [N,32]   leaky(sampled)
#define WS_H       3186688                 // [N,128]  leaky(x@W1+b1)
#define WS_WOUT    4235264                 // [N,32]
#define WS_SCORE   4497408                 // [N]
#define WS_BN1SUM  4505600                 // 32
#define WS_BN1SQ   (WS_BN1SUM + 32)        // 32
#define WS_BN2SUM  (WS_BN1SQ + 32)         // 128
#define WS_BN2SQ   (WS_BN2SUM + 128)       // 128
#define WS_BN1SC   (WS_BN2SQ + 128)        // 32
#define WS_BN1SH   (WS_BN1SC + 32)         // 32
#define WS_BN2SC   (WS_BN1SH + 32)         // 128
#define WS_BN2SH   (WS_BN2SC + 128)        // 128
#define WS_W2P     (WS_BN2SH + 128)        // [128,32] = diag(bn2_scale) @ lin2_W
#define WS_B2P     (WS_W2P + 4096)         // 32       folded bias

__device__ __forceinline__ float leaky_f(float x) {
  return x > 0.f ? x : NEG_SLOPE * x;
}

// fp32 WMMA: D[16x16] += A[16x4] * B[4x16]
// A frag (per lane): rows M = lane&15, K = 2*(lane>>4) + {0,1}
// B frag (per lane): cols N = lane&15, K = 2*(lane>>4) + {0,1}
// C/D (per lane):    vgpr v -> M = v + 8*(lane>>4), N = lane&15
__device__ __forceinline__ v8f wmma4(v2f a, v2f b, v8f c) {
  return __builtin_amdgcn_wmma_f32_16x16x4_f32(false, a, false, b,
                                               (short)0, c, false, false);
}

__device__ __forceinline__ v8f zero8() {
  v8f z = {0.f, 0.f, 0.f, 0.f, 0.f, 0.f, 0.f, 0.f};
  return z;
}

// ---------------------------------------------------------------------------
// gfx1250 async global->LDS copy (VGPR-free staging, tracked by ASYNCcnt).
// Builtin signature (from the round-2 diagnostic): the b128 variant takes
// (v4i addrspace(1)* gsrc, v4i addrspace(3)* ldst, imm offset, imm cpol).
// AS3 pointer = low 32 bits of the flat LDS address (aperture rule);
// AS1 pointer = same virtual address.
// ---------------------------------------------------------------------------
#if __has_builtin(__builtin_amdgcn_global_load_async_to_lds_b128)
#define HAVE_ASYNC_LDS 1
#else
#define HAVE_ASYNC_LDS 0
#endif

typedef __attribute__((address_space(1))) v4i* gptr128;
typedef __attribute__((address_space(3))) v4i* lptr128;

template <int OFF>
__device__ __forceinline__ void async_cp16(const void* g, void* l) {
#if HAVE_ASYNC_LDS
  __builtin_amdgcn_global_load_async_to_lds_b128(
      (gptr128)(unsigned long long)(uintptr_t)g,
      (lptr128)(unsigned int)(uintptr_t)l, OFF, 0);
#else
  *(v4f*)((char*)l + OFF) = *(const v4f*)((const char*)g + OFF);
#endif
}

template <int N>
__device__ __forceinline__ void wait_async() {
#if HAVE_ASYNC_LDS
#if __has_builtin(__builtin_amdgcn_s_wait_asynccnt)
  __builtin_amdgcn_s_wait_asynccnt(N);
#else
  asm volatile("s_wait_asynccnt %0" ::"n"(N) : "memory");
#endif
#endif
}

// ---------------------------------------------------------------------------
// K0: packed geometry (cx,cy,|q|^2), |pos|^2, zero BN-stat accumulators
// ---------------------------------------------------------------------------
__global__ void k_prep(const float* __restrict__ pos,
                       const float* __restrict__ kpos,
                       float* __restrict__ ws) {
  int tid = blockIdx.x * blockDim.x + threadIdx.x;   // grid covers B*Q exactly
  if (tid < B_ * Q_) {
    int b  = tid / Q_;
    int q  = tid - b * Q_;
    int nn = q / KF_;
    int kk = q - nn * KF_;
    float px = pos[(b * NPG_ + nn) * 2 + 0];
    float py = pos[(b * NPG_ + nn) * 2 + 1];
    float cx = px + kpos[kk * 2 + 0];
    float cy = py + kpos[kk * 2 + 1];
    v4f g = {cx, cy, cx * cx + cy * cy, 0.f};
    *(v4f*)(ws + WS_GEO + (long)tid * 4) = g;
  }
  if (tid < NT_) {
    float px = pos[tid * 2 + 0];
    float py = pos[tid * 2 + 1];
    ws[WS_SP2 + tid] = px * px + py * py;
  }
  if (tid < 320) ws[WS_BN1SUM + tid] = 0.f;          // bn1(64) + bn2(256) sums
}

// ---------------------------------------------------------------------------
// K1: conv_w[(b,nn,kk), d] = sum_c w[(b,nn),c] * conv_W[kk,c,d]   (WMMA fp32)
//     grid (64, 9), block 256 (8 waves, one 16x32 output tile each)
// ---------------------------------------------------------------------------
__global__ void __launch_bounds__(256)
k_convw(const float* __restrict__ w, const float* __restrict__ convW,
        float* __restrict__ ws) {
  int l  = threadIdx.x & 31;
  int hh = l >> 4;                 // lane-half
  int lm = l & 15;
  int mtile = blockIdx.x * 8 + (threadIdx.x >> 5);   // 0..511
  int kk = blockIdx.y;
  const float* wrow = w + (mtile * 16 + lm) * C_;    // A rows
  const float* Wk   = convW + kk * C_ * C_;
  v8f acc0 = zero8(), acc1 = zero8();
  #pragma unroll
  for (int ks = 0; ks < C_; ks += 4) {
    int k0 = ks + 2 * hh;
    v2f a;  a.x = wrow[k0];            a.y = wrow[k0 + 1];
    v2f vb0, vb1;
    vb0.x = Wk[k0 * C_ + lm];          vb0.y = Wk[(k0 + 1) * C_ + lm];
    vb1.x = Wk[k0 * C_ + 16 + lm];     vb1.y = Wk[(k0 + 1) * C_ + 16 + lm];
    acc0 = wmma4(a, vb0, acc0);
    acc1 = wmma4(a, vb1, acc1);
  }
  #pragma unroll
  for (int v = 0; v < 8; ++v) {
    int m  = mtile * 16 + v + 8 * hh;                // global (b,nn) row
    int bb = m >> 10, nn = m & (NPG_ - 1);
    long orow = (long)(bb * Q_ + nn * KF_ + kk) * C_;
    ws[WS_CONVW + orow + lm]      = acc0[v];
    ws[WS_CONVW + orow + 16 + lm] = acc1[v];
  }
}

// ---------------------------------------------------------------------------
// K2: fused Gaussian-kernel GEMM with double-buffered async LDS staging:
//     xl[m,c] = leaky( sum_q exp(-0.5*|p_m - q|^2) * conv_w[q,c] )
//     K is never materialized: exp() feeds the fp32 WMMA A-fragment directly.
//     block = 128 (4 waves share a 128-q LDS chunk), grid = B*16
// ---------------------------------------------------------------------------
#define QCH 128
#define CWS 40     // LDS row stride: rows 2 apart land on disjoint bank halves
__global__ void __launch_bounds__(128)
k_sample(const float* __restrict__ pos, float* __restrict__ ws) {
  __shared__ float s_cw[2][QCH][CWS];
  __shared__ float s_geo[2][QCH][4];
  int t  = threadIdx.x;
  int l  = t & 31, hh = l >> 4, lm = l & 15;
  int b      = blockIdx.x >> 4;
  int mtile  = (blockIdx.x & 15) * 4 + (t >> 5);     // 0..63 within graph
  int mbase  = mtile * 16;
  int mrow   = mbase + lm;                            // A row for this lane
  float px   = pos[(b * NPG_ + mrow) * 2 + 0];
  float py   = pos[(b * NPG_ + mrow) * 2 + 1];
  float sp2m = ws[WS_SP2 + b * NPG_ + mrow];
  const float* cw  = ws + WS_CONVW + (long)b * Q_ * C_;
  const float* geo = ws + WS_GEO   + (long)b * Q_ * 4;
  v8f acc0 = zero8(), acc1 = zero8();

  // stage one q-row: 8x b128 (conv_w) + 1x b128 (geometry) = 9 async ops
  auto stage = [&](int chunk, int buf) {
    const float* gc = cw  + (long)(chunk * QCH + t) * C_;
    const float* gg = geo + (long)(chunk * QCH + t) * 4;
    float* lc = &s_cw[buf][t][0];
    async_cp16<0>(gc, lc);    async_cp16<16>(gc, lc);
    async_cp16<32>(gc, lc);   async_cp16<48>(gc, lc);
    async_cp16<64>(gc, lc);   async_cp16<80>(gc, lc);
    async_cp16<96>(gc, lc);   async_cp16<112>(gc, lc);
    async_cp16<0>(gg, &s_geo[buf][t][0]);
  };

  const int NCH = Q_ / QCH;                           // 72 chunks
  stage(0, 0);
  for (int i = 0; i < NCH; ++i) {
    int cur = i & 1;
    if (i + 1 < NCH) {
      stage(i + 1, cur ^ 1);                          // overlap next chunk
      wait_async<9>();                                // chunk i fully landed
    } else {
      wait_async<0>();
    }
    __syncthreads();                                  // LDS visible to all waves

    float (*cwb)[CWS] = s_cw[cur];
    float (*geob)[4]  = s_geo[cur];
    #pragma unroll 4
    for (int qs = 0; qs < QCH; qs += 4) {
      int q0 = qs + 2 * hh;
      float d20 = sp2m + geob[q0][2]
                - 2.f * (px * geob[q0][0] + py * geob[q0][1]);
      float d21 = sp2m + geob[q0 + 1][2]
                - 2.f * (px * geob[q0 + 1][0] + py * geob[q0 + 1][1]);
      v2f a;  a.x = __expf(-0.5f * d20);  a.y = __expf(-0.5f * d21);
      v2f vb0, vb1;
      vb0.x = cwb[q0][lm];          vb0.y = cwb[q0 + 1][lm];
      vb1.x = cwb[q0][16 + lm];     vb1.y = cwb[q0 + 1][16 + lm];
      acc0 = wmma4(a, vb0, acc0);
      acc1 = wmma4(a, vb1, acc1);
    }
    __syncthreads();                                  // buffer reuse fence
  }

  // epilogue: leaky + store + BN1 statistics
  float s0 = 0.f, q0s = 0.f, s1 = 0.f, q1s = 0.f;
  #pragma unroll
  for (int v = 0; v < 8; ++v) {
    int m = b * NPG_ + mbase + v + 8 * hh;
    float x0 = leaky_f(acc0[v]);
    float x1 = leaky_f(acc1[v]);
    ws[WS_XL + m * C_ + lm]      = x0;
    ws[WS_XL + m * C_ + 16 + lm] = x1;
    s0 += x0;  q0s += x0 * x0;  s1 += x1;  q1s += x1 * x1;
  }
  s0  += __shfl_xor(s0, 16);   q0s += __shfl_xor(q0s, 16);
  s1  += __shfl_xor(s1, 16);   q1s += __shfl_xor(q1s, 16);
  if (hh == 0) {
    atomicAdd(&ws[WS_BN1SUM + lm], s0);
    atomicAdd(&ws[WS_BN1SQ  + lm], q0s);
  } else {
    atomicAdd(&ws[WS_BN1SUM + 16 + lm], s1);
    atomicAdd(&ws[WS_BN1SQ  + 16 + lm], q1s);
  }
}

// ---------------------------------------------------------------------------
// K3: fold BN1 stats into per-channel scale/shift
// ---------------------------------------------------------------------------
__global__ void k_bn1fin(const float* __restrict__ g, const float* __restrict__ bt,
                         float* __restrict__ ws) {
  int t = threadIdx.x;                                // 32 threads
  float mean = ws[WS_BN1SUM + t] * (1.f / NT_);
  float var  = ws[WS_BN1SQ + t] * (1.f / NT_) - mean * mean;
  float sc   = rsqrtf(var + BN_EPS) * g[t];
  ws[WS_BN1SC + t] = sc;
  ws[WS_BN1SH + t] = bt[t] - mean * sc;
}

// ---------------------------------------------------------------------------
// K4: x = BN1(xl) + weights ;  h = leaky(x@W1 + b1) ; BN2 stats   (WMMA fp32)
//     block 128 = 4 waves (one 32-col group each), grid 512 M-tiles
// ---------------------------------------------------------------------------
__global__ void __launch_bounds__(128)
k_xlin1(const float* __restrict__ wts, const float* __restrict__ W1,
        const float* __restrict__ bias1, float* __restrict__ ws) {
  __shared__ float xs[16][33];
  int t = threadIdx.x;
  int mtile = blockIdx.x;
  for (int e = t; e < 512; e += 128) {
    int r = e >> 5, c = e & 31;
    int m = mtile * 16 + r;
    float xv = ws[WS_BN1SC + c] * ws[WS_XL + m * C_ + c]
             + ws[WS_BN1SH + c] + wts[m * C_ + c];
    xs[r][c] = xv;
    ws[WS_X + m * C_ + c] = xv;
  }
  __syncthreads();

  int l = t & 31, hh = l >> 4, lm = l & 15;
  int cb = (t >> 5) * 32;                             // output column base
  v8f acc0 = zero8(), acc1 = zero8();
  #pragma unroll
  for (int ks = 0; ks < C_; ks += 4) {
    int k0 = ks + 2 * hh;
    v2f a;  a.x = xs[lm][k0];  a.y = xs[lm][k0 + 1];
    v2f vb0, vb1;
    vb0.x = W1[k0 * CM_ + cb + lm];       vb0.y = W1[(k0 + 1) * CM_ + cb + lm];
    vb1.x = W1[k0 * CM_ + cb + 16 + lm];  vb1.y = W1[(k0 + 1) * CM_ + cb + 16 + lm];
    acc0 = wmma4(a, vb0, acc0);
    acc1 = wmma4(a, vb1, acc1);
  }
  float lb0 = bias1[cb + lm], lb1 = bias1[cb + 16 + lm];
  float s0 = 0.f, q0s = 0.f, s1 = 0.f, q1s = 0.f;
  #pragma unroll
  for (int v = 0; v < 8; ++v) {
    int m = mtile * 16 + v + 8 * hh;
    float h0 = leaky_f(acc0[v] + lb0);
    float h1 = leaky_f(acc1[v] + lb1);
    ws[WS_H + m * CM_ + cb + lm]      = h0;
    ws[WS_H + m * CM_ + cb + 16 + lm] = h1;
    s0 += h0;  q0s += h0 * h0;  s1 += h1;  q1s += h1 * h1;
  }
  s0 += __shfl_xor(s0, 16);  q0s += __shfl_xor(q0s, 16);
  s1 += __shfl_xor(s1, 16);  q1s += __shfl_xor(q1s, 16);
  if (hh == 0) {
    atomicAdd(&ws[WS_BN2SUM + cb + lm], s0);
    atomicAdd(&ws[WS_BN2SQ  + cb + lm], q0s);
  } else {
    atomicAdd(&ws[WS_BN2SUM + cb + 16 + lm], s1);
    atomicAdd(&ws[WS_BN2SQ  + cb + 16 + lm], q1s);
  }
}

// ---------------------------------------------------------------------------
// K5: fold BN2 into lin2:  W2' = diag(bn2_scale)@W2,  b2' = bn2_shift@W2 + b2
// ---------------------------------------------------------------------------
__global__ void k_bn2fin(const float* __restrict__ g2, const float* __restrict__ bt2,
                         const float* __restrict__ W2, const float* __restrict__ b2,
                         float* __restrict__ ws) {
  int t = threadIdx.x;                                // 256 threads
  if (t < CM_) {
    float mean = ws[WS_BN2SUM + t] * (1.f / NT_);
    float var  = ws[WS_BN2SQ + t] * (1.f / NT_) - mean * mean;
    float sc   = rsqrtf(var + BN_EPS) * g2[t];
    ws[WS_BN2SC + t] = sc;
    ws[WS_BN2SH + t] = bt2[t] - mean * sc;
  }
  __syncthreads();
  for (int e = t; e < CM_ * C_; e += 256) {
    int j = e >> 5, c = e & 31;
    ws[WS_W2P + e] = ws[WS_BN2SC + j] * W2[j * C_ + c];
  }
  if (t < C_) {
    float acc = b2[t];
    for (int j = 0; j < CM_; ++j) acc += ws[WS_BN2SH + j] * W2[j * C_ + t];
    ws[WS_B2P + t] = acc;
  }
}

// ---------------------------------------------------------------------------
// K6: w_out = x + h@W2' + b2' ; score = (w_out @ p)/||p||    (WMMA fp32)
//     block 256 = 8 waves (one 16x32 tile each), grid 64
// ---------------------------------------------------------------------------
__global__ void __launch_bounds__(256)
k_wout(const float* __restrict__ sp, float* __restrict__ ws) {
  int t = threadIdx.x, l = t & 31, hh = l >> 4, lm = l & 15;
  int mtile = blockIdx.x * 8 + (t >> 5);
  int mbase = mtile * 16;

  float spl = sp[l];                                  // ||select_p||
  float n2 = spl * spl;
  #pragma unroll
  for (int o = 16; o > 0; o >>= 1) n2 += __shfl_xor(n2, o);
  float invn = rsqrtf(n2);

  const float* hrow = ws + WS_H + (long)(mbase + lm) * CM_;
  const float* W2p  = ws + WS_W2P;
  v8f acc0 = zero8(), acc1 = zero8();
  #pragma unroll 8
  for (int ks = 0; ks < CM_; ks += 4) {
    int k0 = ks + 2 * hh;
    v2f a = *(const v2f*)(hrow + k0);
    v2f vb0, vb1;
    vb0.x = W2p[k0 * C_ + lm];        vb0.y = W2p[(k0 + 1) * C_ + lm];
    vb1.x = W2p[k0 * C_ + 16 + lm];   vb1.y = W2p[(k0 + 1) * C_ + 16 + lm];
    acc0 = wmma4(a, vb0, acc0);
    acc1 = wmma4(a, vb1, acc1);
  }
  float sp0 = sp[lm], sp1 = sp[16 + lm];
  float bz0 = ws[WS_B2P + lm], bz1 = ws[WS_B2P + 16 + lm];
  #pragma unroll
  for (int v = 0; v < 8; ++v) {
    int m = mbase + v + 8 * hh;
    float w0 = ws[WS_X + m * C_ + lm]      + acc0[v] + bz0;
    float w1 = ws[WS_X + m * C_ + 16 + lm] + acc1[v] + bz1;
    ws[WS_WOUT + m * C_ + lm]      = w0;
    ws[WS_WOUT + m * C_ + 16 + lm] = w1;
    float p = w0 * sp0 + w1 * sp1;                    // row dot within half
    #pragma unroll
    for (int o = 1; o < 16; o <<= 1) p += __shfl_xor(p, o);
    if (lm == 0) ws[WS_SCORE + m] = p * invn;
  }
}

// ---------------------------------------------------------------------------
// K7: per-graph bitonic top-512 (lax.top_k order: desc value, asc index),
//     gather positions + tanh-scaled weights, write out_batch
// ---------------------------------------------------------------------------
__global__ void __launch_bounds__(512)
k_topk(const float* __restrict__ pos, float* __restrict__ out,
       float* __restrict__ ws) {
  __shared__ float sc[1024];
  __shared__ int   si[1024];
  int b = blockIdx.x, t = threadIdx.x;
  sc[t]       = ws[WS_SCORE + b * NPG_ + t];        si[t]       = t;
  sc[t + 512] = ws[WS_SCORE + b * NPG_ + t + 512];  si[t + 512] = t + 512;
  for (int kk = 2; kk <= 1024; kk <<= 1) {
    for (int jj = kk >> 1; jj > 0; jj >>= 1) {
      __syncthreads();
      int i  = ((t & ~(jj - 1)) << 1) | (t & (jj - 1));
      int ix = i | jj;
      float s0v = sc[i], s1v = sc[ix];
      int   i0v = si[i], i1v = si[ix];
      bool firstLess = (s0v < s1v) || (s0v == s1v && i0v > i1v);
      bool desc = (i & kk) == 0;
      if (desc == firstLess) {
        sc[i] = s1v;  sc[ix] = s0v;  si[i] = i1v;  si[ix] = i0v;
      }
    }
  }
  __syncthreads();
  int   idx = si[t];
  float th  = tanhf(sc[t]);
  int orow  = b * KSEL_ + t;
  out[orow * 2 + 0] = pos[(b * NPG_ + idx) * 2 + 0];
  out[orow * 2 + 1] = pos[(b * NPG_ + idx) * 2 + 1];
  float* ow = out + B_ * KSEL_ * 2;                  // after pos_sel block
  for (int c = 0; c < C_; ++c)
    ow[orow * C_ + c] = ws[WS_WOUT + (b * NPG_ + idx) * C_ + c] * th;
  if (t == 0)
    ((int*)out)[B_ * KSEL_ * 2 + B_ * KSEL_ * C_ + b] = KSEL_;
}

// ---------------------------------------------------------------------------
extern "C" void kernel_launch(void* const* d_in, const int* in_sizes, int n_in,
                              void* d_out, int out_size, void* d_ws, size_t ws_size,
                              hipStream_t stream) {
  const float* positions = (const float*)d_in[0];
  const float* weights   = (const float*)d_in[1];
  // d_in[2] (batch, int32) is constant n per graph — unused
  const float* kpos  = (const float*)d_in[3];
  const float* convW = (const float*)d_in[4];
  const float* bn1g  = (const float*)d_in[5];
  const float* bn1b  = (const float*)d_in[6];
  const float* W1    = (const float*)d_in[7];
  const float* b1    = (const float*)d_in[8];
  const float* bn2g  = (const float*)d_in[9];
  const float* bn2b  = (const float*)d_in[10];
  const float* W2    = (const float*)d_in[11];
  const float* b2    = (const float*)d_in[12];
  const float* selp  = (const float*)d_in[13];
  float* ws  = (float*)d_ws;
  float* out = (float*)d_out;
  (void)in_sizes; (void)n_in; (void)out_size; (void)ws_size;

  k_prep  <<<(B_ * Q_) / 256, 256, 0, stream>>>(positions, kpos, ws);
  k_convw <<<dim3(64, KF_), 256, 0, stream>>>(weights, convW, ws);
  k_sample<<<B_ * 16, 128, 0, stream>>>(positions, ws);
  k_bn1fin<<<1, 32, 0, stream>>>(bn1g, bn1b, ws);
  k_xlin1 <<<NT_ / 16, 128, 0, stream>>>(weights, W1, b1, ws);
  k_bn2fin<<<1, 256, 0, stream>>>(bn2g, bn2b, W2, b2, ws);
  k_wout  <<<NT_ / 16 / 8, 256, 0, stream>>>(selp, ws);
  k_topk  <<<B_, 512, 0, stream>>>(positions, out, ws);
}